// UNet_DS_t2_88321707475386
// MI455X (gfx1250) — compile-verified
//
#include <hip/hip_runtime.h>
#include <hip/hip_bf16.h>

// DS evidential fusion for MI455X (gfx1250), single-pass, bandwidth-bound.
// xw GEMM done in exact f32 via V_WMMA_F32_16X16X4_F32.

typedef float v2f __attribute__((ext_vector_type(2)));
typedef float v8f __attribute__((ext_vector_type(8)));

#define C_DIM   16
#define K_DIM   20          // prototypes
#define KP      32          // padded prototypes (2 WMMA tiles of 16)
#define M_DIM   4           // classes
#define HWD     262144      // 64*64*64
#define HWD_LOG 18

__global__ __launch_bounds__(256) void ds_evid_kernel(
    const float* __restrict__ x,      // [B,16,64,64,64]
    const float* __restrict__ Wm,     // [20,16]
    const float* __restrict__ BETA,   // [20,4]
    const float* __restrict__ alpha,  // [20,1]
    const float* __restrict__ gamma,  // [20,1]
    float* __restrict__ out)          // [B,5,64,64,64]
{
    __shared__ float s_alphap[KP];
    __shared__ float s_g2[KP];
    __shared__ float s_hw2[KP];       // 0.5*||W_k||^2
    __shared__ float s_U[KP][M_DIM];

    const int tid = threadIdx.x;

    // ---- per-block prototype constants (padded protos -> alphap=0 => identity mass) ----
    if (tid < KP) {
        const int k = tid;
        float ap = 0.f, g2 = 0.f, hw2 = 0.f;
        float U0 = 0.f, U1 = 0.f, U2 = 0.f, U3 = 0.f;
        if (k < K_DIM) {
            float b0 = BETA[k * M_DIM + 0], b1 = BETA[k * M_DIM + 1];
            float b2 = BETA[k * M_DIM + 2], b3 = BETA[k * M_DIM + 3];
            float q0 = b0 * b0, q1 = b1 * b1, q2 = b2 * b2, q3 = b3 * b3;
            float inv = 1.f / (q0 + q1 + q2 + q3);
            U0 = q0 * inv; U1 = q1 * inv; U2 = q2 * inv; U3 = q3 * inv;
            ap = 0.99f / (1.f + __expf(-alpha[k]));
            float g = gamma[k];
            g2 = g * g;
            float w2 = 0.f;
            #pragma unroll
            for (int c = 0; c < C_DIM; ++c) {
                float w = Wm[k * C_DIM + c];
                w2 = fmaf(w, w, w2);
            }
            hw2 = 0.5f * w2;
        }
        s_alphap[k] = ap; s_g2[k] = g2; s_hw2[k] = hw2;
        s_U[k][0] = U0; s_U[k][1] = U1; s_U[k][2] = U2; s_U[k][3] = U3;
    }
    __syncthreads();

    const int lane = tid & 31;
    const int wave = tid >> 5;
    const int half = lane >> 4;      // 0: protos m=0..7 / 16..23 ; 1: m=8..15 / 24..31
    const int ln   = lane & 15;      // point-within-tile (= N column of WMMA D)

    // 16 consecutive spatial points per wave
    const int tile = blockIdx.x * 8 + wave;
    const int s0   = tile << 4;
    const int b    = s0 >> HWD_LOG;
    const int r    = s0 & (HWD - 1);

    const float* xb = x + (size_t)b * (size_t)C_DIM * (size_t)HWD + (size_t)r + (size_t)ln;

    // ---- load A (W rows, uniform) and B (x tile) operands; accumulate partial x^2 ----
    v2f a0[4], a1[4], bm[4];
    float part = 0.f;
    const float* wr0 = Wm + ln * C_DIM;            // proto row ln   (tile 0)
    const float* wr1 = Wm + (16 + ln) * C_DIM;     // proto row 16+ln (tile 1, valid if ln<4)
    #pragma unroll
    for (int t = 0; t < 4; ++t) {
        const int c0 = 4 * t + 2 * half;           // K-pair owned by this half-wave
        float x0 = xb[(size_t)c0 * HWD];
        float x1 = xb[(size_t)(c0 + 1) * HWD];
        v2f bv; bv.x = x0; bv.y = x1; bm[t] = bv;
        part = fmaf(x0, x0, part);
        part = fmaf(x1, x1, part);
        v2f av; av.x = wr0[c0]; av.y = wr0[c0 + 1]; a0[t] = av;
        float w0 = 0.f, w1 = 0.f;
        if (ln < 4) { w0 = wr1[c0]; w1 = wr1[c0 + 1]; }
        v2f aw; aw.x = w0; aw.y = w1; a1[t] = aw;
    }
    // complementary half-waves hold the other 8 channels of the same point
    const float x2 = part + __shfl_xor(part, 16, 32);

    // ---- xw via chained f32 WMMA: D[proto, point] += A[proto,4ch] x B[4ch,point] ----
    v8f acc0 = {}; // protos 0..15
    v8f acc1 = {}; // protos 16..31 (16..19 real)
    #pragma unroll
    for (int t = 0; t < 4; ++t) {
        acc0 = __builtin_amdgcn_wmma_f32_16x16x4_f32(false, a0[t], false, bm[t],
                                                     (short)0, acc0, false, false);
        acc1 = __builtin_amdgcn_wmma_f32_16x16x4_f32(false, a1[t], false, bm[t],
                                                     (short)0, acc1, false, false);
    }

    // ---- per-lane Dempster combination over this lane's 16 prototypes ----
    float f0 = 0.f, f1 = 0.f, f2 = 0.f, f3 = 0.f, o = 1.f;
    const float hx2 = 0.5f * x2;
    const int kbase = 8 * half;
    #pragma unroll
    for (int j = 0; j < 8; ++j) {
        // tile 0 proto
        {
            const int k = kbase + j;
            const float d  = hx2 + s_hw2[k] - acc0[j];
            const float s  = s_alphap[k] * __expf(-s_g2[k] * d);
            const float mo = 1.f - s;
            const float mf0 = s_U[k][0] * s, mf1 = s_U[k][1] * s;
            const float mf2 = s_U[k][2] * s, mf3 = s_U[k][3] * s;
            f0 = fmaf(f0, mf0 + mo, mf0 * o);
            f1 = fmaf(f1, mf1 + mo, mf1 * o);
            f2 = fmaf(f2, mf2 + mo, mf2 * o);
            f3 = fmaf(f3, mf3 + mo, mf3 * o);
            o *= mo;
        }
        // tile 1 proto (padded ones have alphap=0 -> s=0 -> identity)
        {
            const int k = 16 + kbase + j;
            const float d  = hx2 + s_hw2[k] - acc1[j];
            const float s  = s_alphap[k] * __expf(-s_g2[k] * d);
            const float mo = 1.f - s;
            const float mf0 = s_U[k][0] * s, mf1 = s_U[k][1] * s;
            const float mf2 = s_U[k][2] * s, mf3 = s_U[k][3] * s;
            f0 = fmaf(f0, mf0 + mo, mf0 * o);
            f1 = fmaf(f1, mf1 + mo, mf1 * o);
            f2 = fmaf(f2, mf2 + mo, mf2 * o);
            f3 = fmaf(f3, mf3 + mo, mf3 * o);
            o *= mo;
        }
    }

    // ---- merge the two half-wave partial combinations (combination is assoc+comm) ----
    const float pf0 = __shfl_xor(f0, 16, 32);
    const float pf1 = __shfl_xor(f1, 16, 32);
    const float pf2 = __shfl_xor(f2, 16, 32);
    const float pf3 = __shfl_xor(f3, 16, 32);
    const float po  = __shfl_xor(o,  16, 32);
    const float F0 = f0 * pf0 + f0 * po + o * pf0;
    const float F1 = f1 * pf1 + f1 * po + o * pf1;
    const float F2 = f2 * pf2 + f2 * po + o * pf2;
    const float F3 = f3 * pf3 + f3 * po + o * pf3;
    const float O  = o * po;

    // ---- normalize and store (one half-wave writes; [B,5,H,W,D] layout) ----
    const float inv = 1.f / (F0 + F1 + F2 + F3 + O);
    if (half == 0) {
        float* ob = out + (size_t)b * 5u * (size_t)HWD + (size_t)r + (size_t)ln;
        ob[0 * (size_t)HWD] = F0 * inv;
        ob[1 * (size_t)HWD] = F1 * inv;
        ob[2 * (size_t)HWD] = F2 * inv;
        ob[3 * (size_t)HWD] = F3 * inv;
        ob[4 * (size_t)HWD] = O  * inv;
    }
}

extern "C" void kernel_launch(void* const* d_in, const int* in_sizes, int n_in,
                              void* d_out, int out_size, void* d_ws, size_t ws_size,
                              hipStream_t stream) {
    (void)in_sizes; (void)n_in; (void)d_ws; (void)ws_size; (void)out_size;
    const float* x     = (const float*)d_in[0];
    const float* Wm    = (const float*)d_in[1];
    const float* BETA  = (const float*)d_in[2];
    const float* alpha = (const float*)d_in[3];
    const float* gamma = (const float*)d_in[4];
    float* out = (float*)d_out;

    // total points = 2*64^3 = 524288 ; 16 points/wave, 8 waves/block -> 4096 blocks
    dim3 grid(4096), block(256);
    hipLaunchKernelGGL(ds_evid_kernel, grid, block, 0, stream,
                       x, Wm, BETA, alpha, gamma, out);
}